// SIAF_9698036154968
// MI455X (gfx1250) — compile-verified
//
#include <hip/hip_runtime.h>

typedef float v2f __attribute__((ext_vector_type(2)));
typedef float v8f __attribute__((ext_vector_type(8)));

#define LDIM 63
#define DDIM 64
#define HID  8
#define NCT  32   // 512 padded columns / 16
#define NKS  16   // 64 padded K / 4

// ---------------------------------------------------------------------------
// Prep: bake tril mask into W1, pad, and emit the B operand directly in
// V_WMMA_F32_16X16X4_F32 B-fragment order:
//   frag[ct][ks][lane] = { Wt[k, n], Wt[k+1, n] }
//   with n = ct*16 + (lane&15), k = ks*4 + (lane>>4)*2
// so the main loop's per-step B load is one coalesced global_load_b64.
// ---------------------------------------------------------------------------
__global__ __launch_bounds__(64) void siaf_prep(const float* __restrict__ W1,
                                                const float* __restrict__ b1,
                                                float* __restrict__ Bfrag,
                                                float* __restrict__ b1p) {
    int idx = blockIdx.x * blockDim.x + threadIdx.x;
    const int NFRAG = NCT * NKS * 32;
    if (idx < NFRAG) {
        int lane = idx & 31;
        int ks = (idx >> 5) & 15;
        int ct = idx >> 9;
        int n = ct * 16 + (lane & 15);
        int l = n >> 3;
        int h = n & 7;
        int k = ks * 4 + ((lane >> 4) << 1);
        float v0 = 0.f, v1 = 0.f;
        if (l < LDIM) {
            // mask: layer l sees only x[:, :l+1]  ->  k <= l
            if (k     <= l) v0 = W1[(l * HID + h) * LDIM + k];
            if (k + 1 <= l) v1 = W1[(l * HID + h) * LDIM + k + 1];
        }
        Bfrag[idx * 2 + 0] = v0;
        Bfrag[idx * 2 + 1] = v1;
    } else if (idx < NFRAG + NCT * 16) {
        int n = idx - NFRAG;
        int l = n >> 3;
        b1p[n] = (l < LDIM) ? b1[l * HID + (n & 7)] : 0.f;
    }
}

__device__ __forceinline__ float tanh_fast(float v) {
    v = fminf(15.f, fmaxf(-15.f, v));
    float e = __expf(2.f * v);
    return (e - 1.f) / (e + 1.f);
}

// ---------------------------------------------------------------------------
// Main: one wave32 per 16-row batch tile.
//   h(16x512) = tanh(x(16x64) * Wt(64x512) + b1)  via f32 WMMA 16x16x4
//   out[.,l,0/1] = h[.,l,:8] . W2[l,0/1,:]        via 8-lane shfl reductions
//   z = x*exp(alpha)+mu ; log_det = sum(alpha)
// ---------------------------------------------------------------------------
__global__ __launch_bounds__(32) void siaf_main(const float* __restrict__ x,
                                                const float* __restrict__ ip,
                                                const float* __restrict__ W2,
                                                const float* __restrict__ b2,
                                                const float* __restrict__ Bfrag,
                                                const float* __restrict__ b1p,
                                                float* __restrict__ z,
                                                float* __restrict__ logdet) {
    __shared__ float mu_s[DDIM * 16];
    __shared__ float al_s[DDIM * 16];

    const int lane = threadIdx.x;           // 0..31
    const int row0 = blockIdx.x * 16;

    // Preload the 16x64 x-tile as 16 WMMA A-fragments (A layout, ISA 7.12.2):
    // lanes 0-15 hold M=lane, VGPR0/1 = K pair {0,1}; lanes 16-31 = K pair {2,3}.
    const float* xr = x + (size_t)(row0 + (lane & 15)) * DDIM + ((lane >> 4) << 1);
    v2f afrag[NKS];
#pragma unroll
    for (int ks = 0; ks < NKS; ++ks)
        afrag[ks] = *(const v2f*)(xr + ks * 4);

    const int half = lane >> 4;             // which 8-row half of M
    const int hidx = lane & 7;              // h index within an l block
    const int lsel = (lane >> 3) & 1;       // which of the 2 l blocks in this tile

    for (int ct = 0; ct < NCT; ++ct) {
        v8f acc = {0.f, 0.f, 0.f, 0.f, 0.f, 0.f, 0.f, 0.f};
        const float* bp = Bfrag + ((size_t)(ct * NKS) * 32 + lane) * 2;
#pragma unroll
        for (int ks = 0; ks < NKS; ++ks) {
            v2f bfrag = *(const v2f*)(bp + (size_t)ks * 64);
            acc = __builtin_amdgcn_wmma_f32_16x16x4_f32(
                false, afrag[ks], false, bfrag, (short)0, acc, false, false);
        }

        // bias + tanh (column index = ct*16 + (lane&15) for every acc VGPR)
        float bias = b1p[ct * 16 + (lane & 15)];
        float hv[8];
#pragma unroll
        for (int j = 0; j < 8; ++j)
            hv[j] = tanh_fast(acc[j] + bias);

        // second (block-diagonal) contraction: 8-wide dots per (m, l)
        int l = ct * 2 + lsel;
        float w2m = 0.f, w2a = 0.f, b2m = 0.f, b2a = 0.f;
        if (l < LDIM) {
            w2m = W2[(l * 2 + 0) * HID + hidx];
            w2a = W2[(l * 2 + 1) * HID + hidx];
            b2m = b2[l * 2 + 0];
            b2a = b2[l * 2 + 1];
        }
#pragma unroll
        for (int j = 0; j < 8; ++j) {
            float pm = hv[j] * w2m;
            float pa = hv[j] * w2a;
            pm += __shfl_xor(pm, 1, 32);
            pm += __shfl_xor(pm, 2, 32);
            pm += __shfl_xor(pm, 4, 32);
            pa += __shfl_xor(pa, 1, 32);
            pa += __shfl_xor(pa, 2, 32);
            pa += __shfl_xor(pa, 4, 32);
            if (hidx == 0 && l < LDIM) {
                int m = j + (half << 3);        // batch row within tile
                mu_s[l * 16 + m] = pm + b2m;
                al_s[l * 16 + m] = pa + b2a;
            }
        }
    }
    __syncthreads();   // single-wave WG: cheap; forces DScnt drain before reads

    // Epilogue: lane covers dims d = 2*lane, 2*lane+1 for each of the 16 rows.
    const float ip0 = ip[0];
    const float ip1 = ip[1];
    const int d0 = lane * 2;
#pragma unroll 4
    for (int m = 0; m < 16; ++m) {
        float mu0, al0;
        if (lane == 0) { mu0 = ip0; al0 = ip1; }                 // d = 0
        else { mu0 = mu_s[(d0 - 1) * 16 + m]; al0 = al_s[(d0 - 1) * 16 + m]; }
        float mu1 = mu_s[d0 * 16 + m];                           // d = d0+1 -> l = d0
        float al1 = al_s[d0 * 16 + m];

        v2f xv = *(const v2f*)(x + (size_t)(row0 + m) * DDIM + d0);
        v2f zv;
        zv.x = xv.x * __expf(al0) + mu0;
        zv.y = xv.y * __expf(al1) + mu1;
        *(v2f*)(z + (size_t)(row0 + m) * DDIM + d0) = zv;

        float part = al0 + al1;
        part += __shfl_xor(part, 1, 32);
        part += __shfl_xor(part, 2, 32);
        part += __shfl_xor(part, 4, 32);
        part += __shfl_xor(part, 8, 32);
        part += __shfl_xor(part, 16, 32);
        if (lane == 0) logdet[row0 + m] = part;
    }
}

extern "C" void kernel_launch(void* const* d_in, const int* in_sizes, int n_in,
                              void* d_out, int out_size, void* d_ws, size_t ws_size,
                              hipStream_t stream) {
    const float* x  = (const float*)d_in[0];
    const float* ip = (const float*)d_in[1];
    const float* W1 = (const float*)d_in[2];
    const float* b1 = (const float*)d_in[3];
    const float* W2 = (const float*)d_in[4];
    const float* b2 = (const float*)d_in[5];

    const int B = in_sizes[0] / DDIM;

    float* ws    = (float*)d_ws;
    float* Bfrag = ws;                       // 32*16*32*2 floats = 128 KB
    float* b1p   = ws + NCT * NKS * 32 * 2;  // 512 floats

    float* z      = (float*)d_out;
    float* logdet = z + (size_t)B * DDIM;

    const int prep_items = NCT * NKS * 32 + NCT * 16;
    siaf_prep<<<(prep_items + 63) / 64, 64, 0, stream>>>(W1, b1, Bfrag, b1p);
    siaf_main<<<B / 16, 32, 0, stream>>>(x, ip, W2, b2, Bfrag, b1p, z, logdet);
}